// MyDepthWiseConv_1340029796546
// MI455X (gfx1250) — compile-verified
//
#include <hip/hip_runtime.h>

// Problem constants (fixed by the reference)
#define NN    4
#define CH    128
#define HH    128
#define WW    128
#define KSZ   5
#define KK    25
#define PADI  2

#define TH    32                // output rows per block (2 per thread, 16 row-pairs)
#define TROWS (TH + 2 * PADI)   // 36 LDS rows (with halo)
#define TCOLS 136               // 132 used (x cols -2..129), padded to 136 floats
#define TILE_FLOATS (TROWS * TCOLS)   // 4896 floats = 1224 float4

typedef float float4v __attribute__((ext_vector_type(4)));

__global__ __launch_bounds__(256)
void dwconv_ksa_kernel(const float* __restrict__ x,
                       const float* __restrict__ ksa,
                       const float* __restrict__ wgt_g,
                       const float* __restrict__ bias,
                       float* __restrict__ out) {
  __shared__ float xt[TROWS][TCOLS];

  const int tx  = threadIdx.x;            // 0..15  (8-col strip)
  const int ty  = threadIdx.y;            // 0..15  (row pair)
  const int tid = ty * 16 + tx;           // 0..255
  const int h0  = blockIdx.x * TH;
  const int c   = blockIdx.y;
  const int n   = blockIdx.z;

  const float* xplane = x + (((size_t)n * CH + c) * HH) * WW;

  // Hoist uniform scalar loads; SMEM latency overlaps the LDS staging below.
  float wk[KK];
#pragma unroll
  for (int k = 0; k < KK; ++k) wk[k] = wgt_g[c * KK + k];
  const float bv = bias[c];

  const int r0 = ty * 2;                  // two output rows per thread
  const int h  = h0 + r0;
  const int w0 = tx * 8;                  // eight output cols per thread
  const float* ksaBase = ksa + (size_t)n * KK * (HH * WW) + h * WW + w0;

  // 1) Zero the tile: 1224 float4 = 4 full rounds of 256 + 200 guarded.
  //    Halo padding stays zero; interior is overwritten after the barrier.
  {
    float4v* f4 = (float4v*)&xt[0][0];
    const float4v z = {0.f, 0.f, 0.f, 0.f};
#pragma unroll
    for (int it = 0; it < 4; ++it) f4[tid + it * 256] = z;
    if (tid < TILE_FLOATS / 4 - 1024) f4[tid + 1024] = z;
  }
  __syncthreads();  // DS stores committed before async-LDS writes begin

  // 2) Async global -> LDS staging of valid interior rows.
  //    36 rows x 64 8-byte segments = 2304 slots = 9 per thread (constant trip).
  //    Each unrolled slot has a wave-uniform row -> scalar boundary branch.
#pragma unroll
  for (int it = 0; it < 9; ++it) {
    const int s   = tid + it * 256;
    const int row = s >> 6;       // 0..35
    const int seg = s & 63;       // 0..63
    const int gh  = h0 - PADI + row;
    if (gh >= 0 && gh < HH) {
      const float* gp = xplane + gh * WW + seg * 2;
      auto g1 = (const __attribute__((address_space(1))) float*)gp;
      auto l3 = (__attribute__((address_space(3))) float*)&xt[row][PADI + seg * 2];
      asm volatile("global_load_async_to_lds_b64 %0, %1, off"
                   :: "v"(l3), "v"(g1) : "memory");
    }
  }
  // Each wave drains its own ASYNCcnt; the barrier then makes all async LDS
  // writes visible to every wave in the workgroup.
  asm volatile("s_wait_asynccnt 0" ::: "memory");
  __syncthreads();

  // Speculative prefetch of the next row band of this x plane (L2 warm).
  if (h0 + TH < HH) {
    __builtin_prefetch(xplane + (h0 + TH) * WW + tid * 16, 0, 0);
  }

  // 3) Preload the 6 x rows x 12 cols this thread needs (aligned b128 loads).
  float xv[6][12];
#pragma unroll
  for (int i = 0; i < 6; ++i) {
    const float4v xa = *(const float4v*)&xt[r0 + i][w0];
    const float4v xb = *(const float4v*)&xt[r0 + i][w0 + 4];
    const float4v xc = *(const float4v*)&xt[r0 + i][w0 + 8];
    xv[i][0] = xa.x; xv[i][1]  = xa.y; xv[i][2]  = xa.z; xv[i][3]  = xa.w;
    xv[i][4] = xb.x; xv[i][5]  = xb.y; xv[i][6]  = xb.z; xv[i][7]  = xb.w;
    xv[i][8] = xc.x; xv[i][9]  = xc.y; xv[i][10] = xc.z; xv[i][11] = xc.w;
  }

  // 4) Compute: 2 rows x 8 cols per thread, 25 taps, 2 VALU ops per tap/output.
  float a0[8], a1[8];
#pragma unroll
  for (int o = 0; o < 8; ++o) { a0[o] = 0.f; a1[o] = 0.f; }

#pragma unroll
  for (int i = 0; i < KSZ; ++i) {
#pragma unroll
    for (int j = 0; j < KSZ; ++j) {
      const int k = i * KSZ + j;
      // kk-plane stride = 64 KB; +16B / +512B / +528B: 24-bit immediate offsets
      const float* kb = ksaBase + (size_t)k * (HH * WW);
      const float4v s00 = *(const float4v*)(kb);            // row h,   cols 0..3
      const float4v s01 = *(const float4v*)(kb + 4);        // row h,   cols 4..7
      const float4v s10 = *(const float4v*)(kb + WW);       // row h+1, cols 0..3
      const float4v s11 = *(const float4v*)(kb + WW + 4);   // row h+1, cols 4..7
      const float w_k = wk[k];
      a0[0] = fmaf(w_k * s00.x, xv[i][j + 0], a0[0]);
      a0[1] = fmaf(w_k * s00.y, xv[i][j + 1], a0[1]);
      a0[2] = fmaf(w_k * s00.z, xv[i][j + 2], a0[2]);
      a0[3] = fmaf(w_k * s00.w, xv[i][j + 3], a0[3]);
      a0[4] = fmaf(w_k * s01.x, xv[i][j + 4], a0[4]);
      a0[5] = fmaf(w_k * s01.y, xv[i][j + 5], a0[5]);
      a0[6] = fmaf(w_k * s01.z, xv[i][j + 6], a0[6]);
      a0[7] = fmaf(w_k * s01.w, xv[i][j + 7], a0[7]);
      a1[0] = fmaf(w_k * s10.x, xv[i + 1][j + 0], a1[0]);
      a1[1] = fmaf(w_k * s10.y, xv[i + 1][j + 1], a1[1]);
      a1[2] = fmaf(w_k * s10.z, xv[i + 1][j + 2], a1[2]);
      a1[3] = fmaf(w_k * s10.w, xv[i + 1][j + 3], a1[3]);
      a1[4] = fmaf(w_k * s11.x, xv[i + 1][j + 4], a1[4]);
      a1[5] = fmaf(w_k * s11.y, xv[i + 1][j + 5], a1[5]);
      a1[6] = fmaf(w_k * s11.z, xv[i + 1][j + 6], a1[6]);
      a1[7] = fmaf(w_k * s11.w, xv[i + 1][j + 7], a1[7]);
    }
  }

  float* obase = out + (((size_t)n * CH + c) * HH + h) * WW + w0;
  const float4v o00 = {a0[0] + bv, a0[1] + bv, a0[2] + bv, a0[3] + bv};
  const float4v o01 = {a0[4] + bv, a0[5] + bv, a0[6] + bv, a0[7] + bv};
  const float4v o10 = {a1[0] + bv, a1[1] + bv, a1[2] + bv, a1[3] + bv};
  const float4v o11 = {a1[4] + bv, a1[5] + bv, a1[6] + bv, a1[7] + bv};
  *(float4v*)(obase)          = o00;
  *(float4v*)(obase + 4)      = o01;
  *(float4v*)(obase + WW)     = o10;
  *(float4v*)(obase + WW + 4) = o11;
}

extern "C" void kernel_launch(void* const* d_in, const int* in_sizes, int n_in,
                              void* d_out, int out_size, void* d_ws, size_t ws_size,
                              hipStream_t stream) {
  const float* x    = (const float*)d_in[0];   // (N, CH, H, W)
  const float* ksa  = (const float*)d_in[1];   // (N, 25, H, W)
  const float* wgt  = (const float*)d_in[2];   // (CH, 5, 5)
  const float* bias = (const float*)d_in[3];   // (CH)
  float* out = (float*)d_out;                  // (N, CH, H, W)

  dim3 grid(HH / TH, CH, NN);   // 4 x 128 x 4 = 2048 blocks
  dim3 block(16, 16, 1);        // 256 threads = 8 wave32
  hipLaunchKernelGGL(dwconv_ksa_kernel, grid, block, 0, stream,
                     x, ksa, wgt, bias, out);
}